// GNN_NCM_45019847197426
// MI455X (gfx1250) — compile-verified
//
#include <hip/hip_runtime.h>

typedef float v2f __attribute__((ext_vector_type(2)));
typedef float v8f __attribute__((ext_vector_type(8)));

// ---------------------------------------------------------------------------
// Kernel 1: xn = [x | noise]  (N x 36), zero aggr1 (N x 16) and aggr2 (N x 8)
// ---------------------------------------------------------------------------
__global__ void k_prep(const float* __restrict__ x, const float* __restrict__ noise,
                       float* __restrict__ xn, float* __restrict__ aggr1,
                       float* __restrict__ aggr2, int N) {
  int stride = gridDim.x * blockDim.x;
  int t0 = blockIdx.x * blockDim.x + threadIdx.x;
  for (int idx = t0; idx < N * 36; idx += stride) {
    int n = idx / 36, c = idx - n * 36;
    xn[idx] = (c < 32) ? x[(long)n * 32 + c] : noise[(long)n * 4 + (c - 32)];
  }
  for (int idx = t0; idx < N * 16; idx += stride) aggr1[idx] = 0.0f;
  for (int idx = t0; idx < N * 8;  idx += stride) aggr2[idx] = 0.0f;
}

// ---------------------------------------------------------------------------
// Kernel 2: conv1 per-edge MLP.  16 lanes per edge; lane o owns output col o.
// m = relu(xn[src] @ W1[e] + b1[e]) @ W2[e] + b2[e];  atomicAdd -> aggr1[dst]
// Weight reads are single-use: non-temporal so xn stays L2-resident.
// ---------------------------------------------------------------------------
__global__ void k_edge1(const float* __restrict__ xn, const int* __restrict__ ei,
                        const float* __restrict__ w1, const float* __restrict__ b1,
                        const float* __restrict__ w2, const float* __restrict__ b2,
                        float* __restrict__ aggr1, int E) {
  int tid = blockIdx.x * blockDim.x + threadIdx.x;
  int o = tid & 15;
  int e = tid >> 4;
  if (e >= E) return;
  long src = ei[e];
  long dst = ei[E + e];
  const float* xr = xn + src * 36;
  float xa = xr[o];
  float xb = xr[16 + o];
  float xc = (o < 4) ? xr[32 + o] : 0.0f;

  const float* W1 = w1 + (long)e * 576;
  float acc = __builtin_nontemporal_load(&b1[(long)e * 16 + o]);
#pragma unroll
  for (int i = 0; i < 16; ++i)
    acc = fmaf(__shfl(xa, i, 16), __builtin_nontemporal_load(&W1[i * 16 + o]), acc);
#pragma unroll
  for (int i = 0; i < 16; ++i)
    acc = fmaf(__shfl(xb, i, 16), __builtin_nontemporal_load(&W1[(16 + i) * 16 + o]), acc);
#pragma unroll
  for (int i = 0; i < 4; ++i)
    acc = fmaf(__shfl(xc, i, 16), __builtin_nontemporal_load(&W1[(32 + i) * 16 + o]), acc);
  float m = fmaxf(acc, 0.0f);

  const float* W2 = w2 + (long)e * 256;
  float acc2 = __builtin_nontemporal_load(&b2[(long)e * 16 + o]);
#pragma unroll
  for (int i = 0; i < 16; ++i)
    acc2 = fmaf(__shfl(m, i, 16), __builtin_nontemporal_load(&W2[i * 16 + o]), acc2);

  atomicAdd(&aggr1[dst * 16 + o], acc2);
}

// ---------------------------------------------------------------------------
// Kernel 3: conv1 node update via WMMA f32 16x16x4.
// z = [xn | aggr1] (16 nodes x 52) ; u = relu(z@W1+b1) ; h = relu(u@W2+b2)
// One wave per 16-node tile; tile staged in LDS (stride 57 = conflict-free).
// ---------------------------------------------------------------------------
__global__ void __launch_bounds__(128) k_node1(
    const float* __restrict__ xn, const float* __restrict__ aggr1,
    const float* __restrict__ upw1, const float* __restrict__ upb1,
    const float* __restrict__ upw2, const float* __restrict__ upb2,
    float* __restrict__ hout, int N) {
  __shared__ float lds[4][16 * 57];
  int wave = threadIdx.x >> 5;
  int lane = threadIdx.x & 31;
  int tile = blockIdx.x * 4 + wave;
  int row0 = tile * 16;
  if (row0 >= N) return;
  float* zt = &lds[wave][0];

  for (int idx = lane; idx < 16 * 52; idx += 32) {
    int r = idx / 52, c = idx - r * 52;
    int row = row0 + r; if (row >= N) row = N - 1;
    float v = (c < 36) ? xn[(long)row * 36 + c] : aggr1[(long)row * 16 + (c - 36)];
    zt[r * 57 + c] = v;
  }
  asm volatile("s_wait_dscnt 0" ::: "memory");

  int col = lane & 15;
  int kh = (lane >> 4) * 2;     // K sub-offset: 0 for lanes 0-15, 2 for 16-31
  float bias = upb1[col];
  v8f C;
#pragma unroll
  for (int g = 0; g < 8; ++g) C[g] = bias;
#pragma unroll
  for (int k0 = 0; k0 < 52; k0 += 4) {
    v2f A, B;
    A[0] = zt[col * 57 + k0 + kh];
    A[1] = zt[col * 57 + k0 + kh + 1];
    B[0] = upw1[(k0 + kh) * 16 + col];
    B[1] = upw1[(k0 + kh + 1) * 16 + col];
    C = __builtin_amdgcn_wmma_f32_16x16x4_f32(false, A, false, B, (short)0, C, false, false);
  }

  int mrow = (lane >> 4) * 8;   // D fragment: VGPR g holds rows g / g+8
#pragma unroll
  for (int g = 0; g < 8; ++g) zt[(mrow + g) * 57 + col] = fmaxf(C[g], 0.0f);
  asm volatile("s_wait_dscnt 0" ::: "memory");

  float bias2 = upb2[col];
  v8f C2;
#pragma unroll
  for (int g = 0; g < 8; ++g) C2[g] = bias2;
#pragma unroll
  for (int k0 = 0; k0 < 16; k0 += 4) {
    v2f A, B;
    A[0] = zt[col * 57 + k0 + kh];
    A[1] = zt[col * 57 + k0 + kh + 1];
    B[0] = upw2[(k0 + kh) * 16 + col];
    B[1] = upw2[(k0 + kh + 1) * 16 + col];
    C2 = __builtin_amdgcn_wmma_f32_16x16x4_f32(false, A, false, B, (short)0, C2, false, false);
  }
#pragma unroll
  for (int g = 0; g < 8; ++g) {
    int row = row0 + mrow + g;
    if (row < N) hout[(long)row * 16 + col] = fmaxf(C2[g], 0.0f);
  }
}

// ---------------------------------------------------------------------------
// Kernel 4: conv2 per-edge MLP. 8 lanes per edge (4 edges per wave).
// ---------------------------------------------------------------------------
__global__ void k_edge2(const float* __restrict__ h, const int* __restrict__ ei,
                        const float* __restrict__ w1, const float* __restrict__ b1,
                        const float* __restrict__ w2, const float* __restrict__ b2,
                        float* __restrict__ aggr2, int E) {
  int tid = blockIdx.x * blockDim.x + threadIdx.x;
  int o = tid & 7;
  int e = tid >> 3;
  if (e >= E) return;
  long src = ei[e];
  long dst = ei[E + e];
  const float* hr = h + src * 16;
  float xa = hr[o];
  float xb = hr[8 + o];

  const float* W1 = w1 + (long)e * 128;
  float acc = __builtin_nontemporal_load(&b1[(long)e * 8 + o]);
#pragma unroll
  for (int i = 0; i < 8; ++i)
    acc = fmaf(__shfl(xa, i, 8), __builtin_nontemporal_load(&W1[i * 8 + o]), acc);
#pragma unroll
  for (int i = 0; i < 8; ++i)
    acc = fmaf(__shfl(xb, i, 8), __builtin_nontemporal_load(&W1[(8 + i) * 8 + o]), acc);
  float m = fmaxf(acc, 0.0f);

  const float* W2 = w2 + (long)e * 64;
  float acc2 = __builtin_nontemporal_load(&b2[(long)e * 8 + o]);
#pragma unroll
  for (int i = 0; i < 8; ++i)
    acc2 = fmaf(__shfl(m, i, 8), __builtin_nontemporal_load(&W2[i * 8 + o]), acc2);

  atomicAdd(&aggr2[dst * 8 + o], acc2);
}

// ---------------------------------------------------------------------------
// Kernel 5: conv2 node update (WMMA, output padded 8->16 cols) + output head.
// ---------------------------------------------------------------------------
__global__ void __launch_bounds__(128) k_node2(
    const float* __restrict__ h, const float* __restrict__ aggr2,
    const float* __restrict__ upw1, const float* __restrict__ upb1,
    const float* __restrict__ upw2, const float* __restrict__ upb2,
    const float* __restrict__ outw, const float* __restrict__ outb,
    float* __restrict__ out, int N) {
  __shared__ float lds[4][16 * 29];
  int wave = threadIdx.x >> 5;
  int lane = threadIdx.x & 31;
  int tile = blockIdx.x * 4 + wave;
  int row0 = tile * 16;
  if (row0 >= N) return;
  float* zt = &lds[wave][0];

  for (int idx = lane; idx < 16 * 24; idx += 32) {
    int r = idx / 24, c = idx - r * 24;
    int row = row0 + r; if (row >= N) row = N - 1;
    float v = (c < 16) ? h[(long)row * 16 + c] : aggr2[(long)row * 8 + (c - 16)];
    zt[r * 29 + c] = v;
  }
  asm volatile("s_wait_dscnt 0" ::: "memory");

  int col = lane & 15;
  int kh = (lane >> 4) * 2;
  bool act = (col < 8);         // only 8 real output features; pad B with zeros
  float bias = act ? upb1[col] : 0.0f;
  v8f C;
#pragma unroll
  for (int g = 0; g < 8; ++g) C[g] = bias;
#pragma unroll
  for (int k0 = 0; k0 < 24; k0 += 4) {
    v2f A, B;
    A[0] = zt[col * 29 + k0 + kh];
    A[1] = zt[col * 29 + k0 + kh + 1];
    B[0] = act ? upw1[(k0 + kh) * 8 + col] : 0.0f;
    B[1] = act ? upw1[(k0 + kh + 1) * 8 + col] : 0.0f;
    C = __builtin_amdgcn_wmma_f32_16x16x4_f32(false, A, false, B, (short)0, C, false, false);
  }

  int mrow = (lane >> 4) * 8;
#pragma unroll
  for (int g = 0; g < 8; ++g) zt[(mrow + g) * 29 + col] = fmaxf(C[g], 0.0f);
  asm volatile("s_wait_dscnt 0" ::: "memory");

  float bias2 = act ? upb2[col] : 0.0f;
  v8f C2;
#pragma unroll
  for (int g = 0; g < 8; ++g) C2[g] = bias2;
#pragma unroll
  for (int k0 = 0; k0 < 8; k0 += 4) {   // K = 8 (only first 8 u-columns valid)
    v2f A, B;
    A[0] = zt[col * 29 + k0 + kh];
    A[1] = zt[col * 29 + k0 + kh + 1];
    B[0] = act ? upw2[(k0 + kh) * 8 + col] : 0.0f;
    B[1] = act ? upw2[(k0 + kh + 1) * 8 + col] : 0.0f;
    C2 = __builtin_amdgcn_wmma_f32_16x16x4_f32(false, A, false, B, (short)0, C2, false, false);
  }

  // output head: out[node] = relu(h2) . out_w + out_b  (reduce across 16-lane half)
  float ow = act ? outw[col] : 0.0f;
  float ob = outb[0];
#pragma unroll
  for (int g = 0; g < 8; ++g) {
    float v = fmaxf(C2[g], 0.0f) * ow;
    v += __shfl_xor(v, 1, 16);
    v += __shfl_xor(v, 2, 16);
    v += __shfl_xor(v, 4, 16);
    v += __shfl_xor(v, 8, 16);
    int row = row0 + mrow + g;
    if (col == 0 && row < N) out[row] = v + ob;
  }
}

// ---------------------------------------------------------------------------
extern "C" void kernel_launch(void* const* d_in, const int* in_sizes, int n_in,
                              void* d_out, int out_size, void* d_ws, size_t ws_size,
                              hipStream_t stream) {
  const float* x      = (const float*)d_in[0];
  const float* noise  = (const float*)d_in[1];
  const int*   ei     = (const int*)d_in[2];
  const float* c1w1   = (const float*)d_in[3];
  const float* c1b1   = (const float*)d_in[4];
  const float* c1w2   = (const float*)d_in[5];
  const float* c1b2   = (const float*)d_in[6];
  const float* c1uw1  = (const float*)d_in[7];
  const float* c1ub1  = (const float*)d_in[8];
  const float* c1uw2  = (const float*)d_in[9];
  const float* c1ub2  = (const float*)d_in[10];
  const float* c2w1   = (const float*)d_in[11];
  const float* c2b1   = (const float*)d_in[12];
  const float* c2w2   = (const float*)d_in[13];
  const float* c2b2   = (const float*)d_in[14];
  const float* c2uw1  = (const float*)d_in[15];
  const float* c2ub1  = (const float*)d_in[16];
  const float* c2uw2  = (const float*)d_in[17];
  const float* c2ub2  = (const float*)d_in[18];
  const float* outw   = (const float*)d_in[19];
  const float* outb   = (const float*)d_in[20];

  int N = in_sizes[0] / 32;   // x is [N, 32]
  int E = in_sizes[2] / 2;    // edge_index is [2, E]

  float* ws    = (float*)d_ws;
  float* xn    = ws;                          // N*36
  float* aggr1 = xn + (size_t)N * 36;         // N*16
  float* hbuf  = aggr1 + (size_t)N * 16;      // N*16
  float* aggr2 = hbuf + (size_t)N * 16;       // N*8

  k_prep<<<2048, 256, 0, stream>>>(x, noise, xn, aggr1, aggr2, N);

  {
    long threads = (long)E * 16;
    k_edge1<<<(threads + 255) / 256, 256, 0, stream>>>(xn, ei, c1w1, c1b1, c1w2, c1b2, aggr1, E);
  }

  int tiles = (N + 15) / 16;
  k_node1<<<(tiles + 3) / 4, 128, 0, stream>>>(xn, aggr1, c1uw1, c1ub1, c1uw2, c1ub2, hbuf, N);

  {
    long threads = (long)E * 8;
    k_edge2<<<(threads + 255) / 256, 256, 0, stream>>>(hbuf, ei, c2w1, c2b1, c2w2, c2b2, aggr2, E);
  }

  k_node2<<<(tiles + 3) / 4, 128, 0, stream>>>(hbuf, aggr2, c2uw1, c2ub1, c2uw2, c2ub2,
                                               outw, outb, (float*)d_out, N);
}